// Geoformer_41618233098784
// MI455X (gfx1250) — compile-verified
//
#include <hip/hip_runtime.h>
#include <math.h>

// ---------------------------------------------------------------------------
// Geoformer head for MI455X (gfx1250, wave32).
// mean-pool(conv3x3(SAME)) over the FPN is linear, so the entire FPN reduces
// to 9 per-channel stats of the raw features:
//   {total, topRow, botRow, leftCol, rightCol, 4 corners}
// nearest-2x upsample-add combines stats with factors {4^d, 2^d, 1}. The heavy
// 3x3 convs (312 GFLOP) become small fp32 WMMA (16x16x4 f32) GEMMs; the only
// heavy cost is one coalesced streaming read of f0..f3 (~146 MB -> ~6us at
// 23.3 TB/s HBM), vectorized to b128 loads for the two big levels.
// ---------------------------------------------------------------------------

typedef float v2f __attribute__((ext_vector_type(2)));
typedef float v8f __attribute__((ext_vector_type(8)));

#if __has_builtin(__builtin_amdgcn_wmma_f32_16x16x4_f32)
#define WMMA_F32X4 1
#endif

#define B_TOK 64
#define FPN_C 256
#define DIM   1024
#define NE    32
#define TOPK  3
#define HID   256
#define OUTD  64
#define NQ    11399
#define NOUT  (NQ + 2)      // 11401 per-token outputs
#define NSTAT 576           // B_TOK * 9

__device__ __forceinline__ float gelu_exact(float v) {
    return 0.5f * v * (1.0f + erff(v * 0.70710678118654752440f));
}

// ---------------- shared LDS reduction for the 5 running sums ---------------
__device__ __forceinline__ void stats_reduce_store(float tot, float rT, float rB,
                                                   float cL, float cR,
                                                   const float* p, int H, int W,
                                                   float* o) {
    __shared__ float sh[5][256];
    int tid = threadIdx.x;
    sh[0][tid] = tot; sh[1][tid] = rT; sh[2][tid] = rB; sh[3][tid] = cL; sh[4][tid] = cR;
    __syncthreads();
    for (int s = 128; s > 0; s >>= 1) {
        if (tid < s)
            for (int v = 0; v < 5; ++v) sh[v][tid] += sh[v][tid + s];
        __syncthreads();
    }
    if (tid == 0) {
        o[0] = sh[0][0]; o[1] = sh[1][0]; o[2] = sh[2][0]; o[3] = sh[3][0]; o[4] = sh[4][0];
        o[5] = p[0];                      // c00
        o[6] = p[W - 1];                  // c0W
        o[7] = p[(size_t)(H - 1) * W];    // cH0
        o[8] = p[(size_t)H * W - 1];      // cHW
    }
}

// ---------- stats, vectorized b128 path (requires W % 4 == 0) ---------------
__global__ void k_stats_vec(const float* __restrict__ f, float* __restrict__ SF,
                            int C, int H, int W) {
    int b = blockIdx.x / C, c = blockIdx.x % C;
    const float* p = f + (size_t)(b * C + c) * (H * W);
    const float4* p4 = reinterpret_cast<const float4*>(p);
    int tid = threadIdx.x;
    int Wq = W >> 2;            // chunks per row; chunks never straddle rows
    int nvec = H * Wq;
    float tot = 0.f, rT = 0.f, rB = 0.f, cL = 0.f, cR = 0.f;
    for (int i = tid; i < nvec; i += 256) {
        int y = i / Wq, xq = i - y * Wq;
        float4 v = p4[i];
        float s = (v.x + v.y) + (v.z + v.w);
        tot += s;
        if (y == 0)       rT += s;
        if (y == H - 1)   rB += s;
        if (xq == 0)      cL += v.x;
        if (xq == Wq - 1) cR += v.w;
    }
    stats_reduce_store(tot, rT, rB, cL, cR, p, H, W,
                       SF + (size_t)c * NSTAT + b * 9);
}

// ------------------------ stats, scalar path (small levels) -----------------
__global__ void k_stats(const float* __restrict__ f, float* __restrict__ SF,
                        int C, int H, int W) {
    int b = blockIdx.x / C, c = blockIdx.x % C;
    const float* p = f + (size_t)(b * C + c) * (H * W);
    int tid = threadIdx.x;
    float tot = 0.f, rT = 0.f, rB = 0.f, cL = 0.f, cR = 0.f;
    int HW = H * W;
    for (int i = tid; i < HW; i += 256) {
        int y = i / W, xx = i - y * W;
        float v = p[i];
        tot += v;
        if (y == 0)      rT += v;
        if (y == H - 1)  rB += v;
        if (xx == 0)     cL += v;
        if (xx == W - 1) cR += v;
    }
    stats_reduce_store(tot, rT, rB, cL, cR, p, H, W,
                       SF + (size_t)c * NSTAT + b * 9);
}

// ------------- GEMM1: statsLat[256 x 576] = lat_w[256 x C] * statsF[C x 576] -
__global__ void k_gemm1(const float* __restrict__ Wl, const float* __restrict__ SF,
                        float* __restrict__ SL, int C) {
    int lane = threadIdx.x & 31, wv = threadIdx.x >> 5;
    int t = blockIdx.x * 8 + wv;          // 0..575  (16 M-tiles x 36 N-tiles)
    int tM = t & 15, tN = t >> 4;
    int mcol = lane & 15, half = lane >> 4;
    v8f acc = {0.f,0.f,0.f,0.f,0.f,0.f,0.f,0.f};
#ifdef WMMA_F32X4
    int k0 = half * 2;
    for (int kk = 0; kk < C; kk += 4) {
        v2f a, bb;
        a.x  = Wl[(tM * 16 + mcol) * C + kk + k0];
        a.y  = Wl[(tM * 16 + mcol) * C + kk + k0 + 1];
        bb.x = SF[(size_t)(kk + k0) * NSTAT + tN * 16 + mcol];
        bb.y = SF[(size_t)(kk + k0 + 1) * NSTAT + tN * 16 + mcol];
        acc = __builtin_amdgcn_wmma_f32_16x16x4_f32(false, a, false, bb, (short)0, acc,
                                                    false, false);
    }
#else
    for (int kk = 0; kk < C; kk += 4)
        for (int r = 0; r < 8; ++r) {
            int row = tM * 16 + r + half * 8;
            for (int k4 = 0; k4 < 4; ++k4)
                acc[r] += Wl[row * C + kk + k4] * SF[(size_t)(kk + k4) * NSTAT + tN * 16 + mcol];
        }
#endif
    for (int r = 0; r < 8; ++r) {
        int row = tM * 16 + r + half * 8;
        SL[(size_t)row * NSTAT + tN * 16 + mcol] = acc[r];
    }
}

// --------- top-down combine of stats + convert 9 stats -> 9 conv taps -------
__global__ void k_combine(const float* SL0, const float* SL1, const float* SL2, const float* SL3,
                          const float* lb0, const float* lb1, const float* lb2, const float* lb3,
                          float* S90, float* S91, float* S92, float* S93) {
    int id = blockIdx.x * blockDim.x + threadIdx.x;   // 4*64*256
    int lvl = id >> 14;
    int rem = id & 16383;
    int b = rem >> 8, oc = rem & 255;
    const float* SLs[4] = {SL0, SL1, SL2, SL3};
    const float* lbs[4] = {lb0, lb1, lb2, lb3};
    float* S9s[4] = {S90, S91, S92, S93};
    const int Hs[4] = {56, 28, 14, 7};
    float st[9];
    for (int s = 0; s < 9; ++s) st[s] = 0.f;
    float off = 0.f;
    for (int j = lvl; j < 4; ++j) {
        int d = j - lvl;
        float fe = (float)(1 << d), ft = fe * fe;
        const float* s = SLs[j] + (size_t)oc * NSTAT + b * 9;
        st[0] += ft * s[0];
        st[1] += fe * s[1]; st[2] += fe * s[2]; st[3] += fe * s[3]; st[4] += fe * s[4];
        st[5] += s[5]; st[6] += s[6]; st[7] += s[7]; st[8] += s[8];
        off += lbs[j][oc];
    }
    float Hf = (float)Hs[lvl];
    st[0] += off * Hf * Hf;
    st[1] += off * Hf; st[2] += off * Hf; st[3] += off * Hf; st[4] += off * Hf;
    st[5] += off; st[6] += off; st[7] += off; st[8] += off;
    float* o = S9s[lvl] + (size_t)b * 2304 + oc * 9;
    for (int dy = -1; dy <= 1; ++dy)
        for (int dx = -1; dx <= 1; ++dx) {
            float v = st[0];
            if (dy == -1) v -= st[2]; else if (dy == 1) v -= st[1];
            if (dx == -1) v -= st[4]; else if (dx == 1) v -= st[3];
            if (dy == -1 && dx == -1) v += st[8];
            if (dy == -1 && dx ==  1) v += st[7];
            if (dy ==  1 && dx == -1) v += st[6];
            if (dy ==  1 && dx ==  1) v += st[5];
            o[(dy + 1) * 3 + (dx + 1)] = v;
        }
}

// --- GEMM2: pooled[64 x 256] = S9[64 x 2304] * fpn_w^T[2304 x 256] (scaled) --
__global__ void k_gemm2(const float* __restrict__ S9, const float* __restrict__ Wf,
                        const float* __restrict__ bf, float* __restrict__ x,
                        int level, float invHW) {
    int lane = threadIdx.x & 31, wv = threadIdx.x >> 5;
    int t = blockIdx.x * 8 + wv;          // 0..63 (4 M-tiles x 16 N-tiles)
    int tM = t & 3, tN = t >> 2;
    int mcol = lane & 15, half = lane >> 4;
    v8f acc = {0.f,0.f,0.f,0.f,0.f,0.f,0.f,0.f};
#ifdef WMMA_F32X4
    int k0 = half * 2;
    for (int kk = 0; kk < 2304; kk += 4) {
        v2f a, bb;
        a.x  = S9[(size_t)(tM * 16 + mcol) * 2304 + kk + k0];
        a.y  = S9[(size_t)(tM * 16 + mcol) * 2304 + kk + k0 + 1];
        bb.x = Wf[(size_t)(tN * 16 + mcol) * 2304 + kk + k0];
        bb.y = Wf[(size_t)(tN * 16 + mcol) * 2304 + kk + k0 + 1];
        acc = __builtin_amdgcn_wmma_f32_16x16x4_f32(false, a, false, bb, (short)0, acc,
                                                    false, false);
    }
#else
    for (int kk = 0; kk < 2304; kk += 4)
        for (int r = 0; r < 8; ++r) {
            int row = tM * 16 + r + half * 8;
            for (int k4 = 0; k4 < 4; ++k4)
                acc[r] += S9[(size_t)row * 2304 + kk + k4] *
                          Wf[(size_t)(tN * 16 + mcol) * 2304 + kk + k4];
        }
#endif
    for (int r = 0; r < 8; ++r) {
        int row = tM * 16 + r + half * 8;   // token b
        int col = tN * 16 + mcol;           // out channel o
        x[(size_t)row * DIM + level * FPN_C + col] = acc[r] * invHW + bf[col];
    }
}

// ----------------- router + shared expert + top-3 routed experts ------------
__global__ void k_moe(const float* __restrict__ x,
                      const float* __restrict__ Wr, const float* __restrict__ br,
                      const float* __restrict__ Ws1, const float* __restrict__ bs1,
                      const float* __restrict__ Ws2, const float* __restrict__ bs2,
                      const float* __restrict__ We1, const float* __restrict__ be1,
                      const float* __restrict__ We2, const float* __restrict__ be2,
                      float* __restrict__ probs_ws, int* __restrict__ topi_ws,
                      float* __restrict__ comb_ws) {
    int b = blockIdx.x, tid = threadIdx.x;
    __shared__ float xs[DIM];
    __shared__ float hs[HID];
    __shared__ float lg[NE];
    __shared__ float pr[NE];
    __shared__ int   ti[TOPK];
    __shared__ float np[TOPK];
    __shared__ float cb[OUTD];
    for (int d = tid; d < DIM; d += 256) xs[d] = x[(size_t)b * DIM + d];
    __syncthreads();
    if (tid < NE) {
        float a = br[tid];
        for (int d = 0; d < DIM; ++d) a += xs[d] * Wr[d * NE + tid];
        lg[tid] = a;
    }
    __syncthreads();
    if (tid == 0) {
        float m = lg[0];
        for (int e = 1; e < NE; ++e) m = fmaxf(m, lg[e]);
        float s = 0.f;
        for (int e = 0; e < NE; ++e) { float p = expf(lg[e] - m); pr[e] = p; s += p; }
        for (int e = 0; e < NE; ++e) pr[e] /= s;
        unsigned used = 0; float tsum = 0.f;
        for (int k = 0; k < TOPK; ++k) {
            int bi = 0; float bv = -1.f;
            for (int e = 0; e < NE; ++e)
                if (!((used >> e) & 1u) && pr[e] > bv) { bv = pr[e]; bi = e; }
            used |= 1u << bi; ti[k] = bi; np[k] = bv; tsum += bv;
        }
        for (int k = 0; k < TOPK; ++k) np[k] /= tsum;
        for (int e = 0; e < NE; ++e) probs_ws[b * NE + e] = pr[e];
        for (int k = 0; k < TOPK; ++k) topi_ws[b * TOPK + k] = ti[k];
    }
    __syncthreads();
    { // shared expert
        float a = bs1[tid];
        for (int d = 0; d < DIM; ++d) a += xs[d] * Ws1[d * HID + tid];
        hs[tid] = gelu_exact(a);
    }
    __syncthreads();
    if (tid < OUTD) {
        float a = bs2[tid];
        for (int h = 0; h < HID; ++h) a += hs[h] * Ws2[h * OUTD + tid];
        cb[tid] = a;
    }
    __syncthreads();
    for (int k = 0; k < TOPK; ++k) {
        int e = ti[k];
        float a = be1[e * HID + tid];
        const float* W1 = We1 + (size_t)e * DIM * HID;
        for (int d = 0; d < DIM; ++d) a += xs[d] * W1[d * HID + tid];
        hs[tid] = gelu_exact(a);
        __syncthreads();
        if (tid < OUTD) {
            float y = be2[e * OUTD + tid];
            const float* W2 = We2 + (size_t)e * HID * OUTD;
            for (int h = 0; h < HID; ++h) y += hs[h] * W2[h * OUTD + tid];
            cb[tid] += np[k] * y;
        }
        __syncthreads();
    }
    if (tid < OUTD) comb_ws[b * OUTD + tid] = cb[tid];
}

// ------------------ output head: [64 x 64] x [64 x 11401] -------------------
__device__ __forceinline__ float head_w(const float* Wll, const float* Wqt, int k, int col) {
    if (col < 2)    return Wll[k * 2 + col];
    if (col < NOUT) return Wqt[(size_t)k * NQ + (col - 2)];
    return 0.f;
}

__global__ void k_head(const float* __restrict__ comb,
                       const float* __restrict__ Wll, const float* __restrict__ bll,
                       const float* __restrict__ Wqt, const float* __restrict__ bqt,
                       float* __restrict__ out) {
    const int tilesN = (NOUT + 15) / 16;  // 713
    int lane = threadIdx.x & 31, wv = threadIdx.x >> 5;
    int t = blockIdx.x * 8 + wv;
    if (t >= 4 * tilesN) return;          // wave-uniform exit
    int tM = t / tilesN, tN = t - tM * tilesN;
    int mcol = lane & 15, half = lane >> 4;
    int col = tN * 16 + mcol;
    v8f acc = {0.f,0.f,0.f,0.f,0.f,0.f,0.f,0.f};
#ifdef WMMA_F32X4
    int k0 = half * 2;
    for (int kk = 0; kk < OUTD; kk += 4) {
        v2f a, bb;
        a.x  = comb[(tM * 16 + mcol) * OUTD + kk + k0];
        a.y  = comb[(tM * 16 + mcol) * OUTD + kk + k0 + 1];
        bb.x = head_w(Wll, Wqt, kk + k0, col);
        bb.y = head_w(Wll, Wqt, kk + k0 + 1, col);
        acc = __builtin_amdgcn_wmma_f32_16x16x4_f32(false, a, false, bb, (short)0, acc,
                                                    false, false);
    }
#else
    for (int kk = 0; kk < OUTD; kk += 4)
        for (int r = 0; r < 8; ++r) {
            int row = tM * 16 + r + half * 8;
            for (int k4 = 0; k4 < 4; ++k4)
                acc[r] += comb[row * OUTD + kk + k4] * head_w(Wll, Wqt, kk + k4, col);
        }
#endif
    float bias = (col < 2) ? bll[col] : ((col < NOUT) ? bqt[col - 2] : 0.f);
    for (int r = 0; r < 8; ++r) {
        int row = tM * 16 + r + half * 8;   // token
        if (col < NOUT) out[(size_t)row * NOUT + col] = acc[r] + bias;
    }
}

// ----------------------------- balance loss ---------------------------------
__global__ void k_bal(const float* __restrict__ probs, const int* __restrict__ topi,
                      float* __restrict__ out) {
    if (threadIdx.x == 0 && blockIdx.x == 0) {
        float bal = 0.f;
        for (int e = 0; e < NE; ++e) {
            int cnt = 0; float ps = 0.f;
            for (int b = 0; b < B_TOK; ++b) {
                ps += probs[b * NE + e];
                for (int k = 0; k < TOPK; ++k)
                    if (topi[b * TOPK + k] == e) ++cnt;
            }
            float f_i = ((float)NE / ((float)TOPK * (float)B_TOK)) * (float)cnt;
            float P_i = ps / (float)B_TOK;
            bal += f_i * P_i;
        }
        out[0] = bal * 0.001f;
    }
}

// ---------------------------------------------------------------------------
extern "C" void kernel_launch(void* const* d_in, const int* in_sizes, int n_in,
                              void* d_out, int out_size, void* d_ws, size_t ws_size,
                              hipStream_t stream) {
    (void)in_sizes; (void)n_in; (void)out_size; (void)ws_size;
    const int CH[4] = {96, 192, 384, 768};
    const int SZ[4] = {56, 28, 14, 7};

    const float* f[4];    for (int i = 0; i < 4; ++i) f[i]    = (const float*)d_in[i];
    const float* latw[4]; for (int i = 0; i < 4; ++i) latw[i] = (const float*)d_in[4 + i];
    const float* latb[4]; for (int i = 0; i < 4; ++i) latb[i] = (const float*)d_in[8 + i];
    const float* fpnw[4]; for (int i = 0; i < 4; ++i) fpnw[i] = (const float*)d_in[12 + i];
    const float* fpnb[4]; for (int i = 0; i < 4; ++i) fpnb[i] = (const float*)d_in[16 + i];
    const float* Wr  = (const float*)d_in[20]; const float* br  = (const float*)d_in[21];
    const float* Ws1 = (const float*)d_in[22]; const float* bs1 = (const float*)d_in[23];
    const float* Ws2 = (const float*)d_in[24]; const float* bs2 = (const float*)d_in[25];
    const float* We1 = (const float*)d_in[26]; const float* be1 = (const float*)d_in[27];
    const float* We2 = (const float*)d_in[28]; const float* be2 = (const float*)d_in[29];
    const float* Wll = (const float*)d_in[30]; const float* bll = (const float*)d_in[31];
    const float* Wqt = (const float*)d_in[32]; const float* bqt = (const float*)d_in[33];
    float* out = (float*)d_out;

    // workspace carve-up (floats); total ~2.1M floats (~8.4 MB)
    float* ws = (float*)d_ws;
    size_t off = 0;
    float* SF[4];
    for (int i = 0; i < 4; ++i) { SF[i] = ws + off; off += (size_t)CH[i] * NSTAT; }
    float* SL[4];
    for (int i = 0; i < 4; ++i) { SL[i] = ws + off; off += (size_t)FPN_C * NSTAT; }
    float* S9[4];
    for (int i = 0; i < 4; ++i) { S9[i] = ws + off; off += (size_t)B_TOK * 2304; }
    float* xbuf  = ws + off; off += (size_t)B_TOK * DIM;
    float* probs = ws + off; off += (size_t)B_TOK * NE;
    int*   topi  = (int*)(ws + off); off += (size_t)B_TOK * TOPK;
    float* comb  = ws + off; off += (size_t)B_TOK * OUTD;

    // 1) streaming stats over raw features (the only heavy memory traffic).
    //    b128-vectorized when W % 4 == 0 (levels 0,1 = 79% of bytes).
    for (int i = 0; i < 4; ++i) {
        if ((SZ[i] & 3) == 0)
            k_stats_vec<<<B_TOK * CH[i], 256, 0, stream>>>(f[i], SF[i], CH[i], SZ[i], SZ[i]);
        else
            k_stats<<<B_TOK * CH[i], 256, 0, stream>>>(f[i], SF[i], CH[i], SZ[i], SZ[i]);
    }

    // 2) lateral 1x1 as WMMA GEMM on stats:  [256 x C] * [C x 576]
    for (int i = 0; i < 4; ++i)
        k_gemm1<<<72, 256, 0, stream>>>(latw[i], SF[i], SL[i], CH[i]);

    // 3) top-down combine (+lat bias) and stats->tap conversion
    k_combine<<<256, 256, 0, stream>>>(SL[0], SL[1], SL[2], SL[3],
                                       latb[0], latb[1], latb[2], latb[3],
                                       S9[0], S9[1], S9[2], S9[3]);

    // 4) 3x3-conv + mean-pool as WMMA GEMM:  [64 x 2304] * [2304 x 256]
    for (int i = 0; i < 4; ++i) {
        float invHW = 1.0f / (float)(SZ[i] * SZ[i]);
        k_gemm2<<<8, 256, 0, stream>>>(S9[i], fpnw[i], fpnb[i], xbuf, i, invHW);
    }

    // 5) router softmax/top-3, shared + routed experts
    k_moe<<<B_TOK, 256, 0, stream>>>(xbuf, Wr, br, Ws1, bs1, Ws2, bs2,
                                     We1, be1, We2, be2, probs, topi, comb);

    // 6) output head (lat_lon | qt) via WMMA
    {
        int tiles = 4 * ((NOUT + 15) / 16);         // 2852
        int blocks = (tiles + 7) / 8;               // 357
        k_head<<<blocks, 256, 0, stream>>>(comb, Wll, bll, Wqt, bqt, out);
    }

    // 7) balance loss scalar
    k_bal<<<1, 32, 0, stream>>>(probs, topi, out + (size_t)B_TOK * NOUT);
}